// GAT_61057255080263
// MI455X (gfx1250) — compile-verified
//
#include <hip/hip_runtime.h>
#include <hip/hip_bf16.h>

#define N_NODES 50000
#define N_EDGES 1600000
#define IN_DIM 256
#define HID 64
#define OUTD 16
#define NEG_SLOPE 0.01f

typedef __attribute__((ext_vector_type(2))) float v2f;
typedef __attribute__((ext_vector_type(8))) float v8f;
typedef __attribute__((ext_vector_type(4))) unsigned u32x4;
typedef __attribute__((ext_vector_type(8))) int i32x8;
typedef __attribute__((ext_vector_type(4))) int i32x4;

// ---------------------------------------------------------------------------
// order-preserving float <-> uint map for atomic segment-max
// ---------------------------------------------------------------------------
__device__ __forceinline__ unsigned enc_f(float f) {
    unsigned b = __float_as_uint(f);
    return (b & 0x80000000u) ? ~b : (b | 0x80000000u);
}
__device__ __forceinline__ float dec_f(unsigned u) {
    unsigned b = (u & 0x80000000u) ? (u & 0x7FFFFFFFu) : ~u;
    return __uint_as_float(b);
}

// ---------------------------------------------------------------------------
// zero fill
// ---------------------------------------------------------------------------
__global__ void zero_kernel(float* __restrict__ p, int n) {
    int i = blockIdx.x * blockDim.x + threadIdx.x;
    if (i < n) p[i] = 0.0f;
}

// ---------------------------------------------------------------------------
// GEMM0: Hw0[N,64] = X[N,256] @ W0[256,64], f32 WMMA 16x16x4
// block = 128 threads (4 waves). Each block: one 16-row tile; wave w covers
// output columns [16w, 16w+16). A tile staged in LDS by the Tensor Data
// Mover: 2D D# (tile 256x16 f32, row stride 256), with the TDM's LDS-pad
// feature (pad_interval=7 -> every 256 DWORDs, pad_amount=3 -> +4 DWORDs)
// reproducing the +4-float row pad that kills the 16-way bank conflict on
// the A-fragment ds reads. 50000 = 3125*16 -> every wave has full EXEC.
// ---------------------------------------------------------------------------
#define APAD 260  // 256 + 4 floats row pad (matches TDM pad settings)
__global__ void __launch_bounds__(128) gemm0_wmma(const float* __restrict__ X,
                                                  const float* __restrict__ W,
                                                  float* __restrict__ Hw) {
    __shared__ float As[16 * APAD];

    const int row0 = blockIdx.x * 16;
    const int t = threadIdx.x;

    // hint: pull the (L2-resident) weight matrix in early
    if (t < 64) __builtin_prefetch(W + t * 256, 0, 1);

#if __has_builtin(__builtin_amdgcn_tensor_load_to_lds)
    // --- TDM path: wave 0 issues one tensor DMA for the whole 16x256 tile ---
    if (t < 32) {  // wave-uniform branch (wave 0 only)
        const unsigned lds_base = (unsigned)(unsigned long long)(void*)As;
        const unsigned long long ga =
            (unsigned long long)(const void*)(X + (long)row0 * IN_DIM);
        // D# group 0: count=1 | lds_addr | global_addr[56:0] | type=2
        u32x4 g0;
        g0.x = 1u;                                   // count=1, user mode
        g0.y = lds_base;                             // LDS byte address
        g0.z = (unsigned)(ga & 0xFFFFFFFFu);         // global_addr[31:0]
        g0.w = (unsigned)((ga >> 32) & 0x01FFFFFFu)  // global_addr[56:32]
               | 0x80000000u;                        // type=2 ("image")
        // D# group 1: data_size=4B, pad every 256 DWORDs by 4 DWORDs,
        // tensor_dim0=256, tensor_dim1=16, tile_dim0=256, tile_dim1=16,
        // tensor_dim0_stride=256
        i32x8 g1;
        g1[0] = (int)((2u << 16)      // data_size: 4 bytes
                    | (1u << 20)      // pad_enable
                    | (7u << 22)      // pad_interval: 256 DWORDs
                    | (3u << 25));    // pad_amount: 4 DWORDs (16B)
        g1[1] = (int)(256u << 16);    // [47:32]=0 barrier addr | tensor_dim0 lo16
        g1[2] = (int)(16u << 16);     // tensor_dim0 hi16=0 | tensor_dim1 lo16=16
        g1[3] = (int)(256u << 16);    // tensor_dim1 hi16=0 | tile_dim0=256
        g1[4] = 16;                   // tile_dim1=16 | tile_dim2=0
        g1[5] = 256;                  // tensor_dim0_stride[31:0]
        g1[6] = 0;                    // stride hi | tensor_dim1_stride lo
        g1[7] = 0;                    // tensor_dim1_stride hi
        i32x4 gz4 = {0, 0, 0, 0};                 // groups 2/3 unused (2D)
        i32x8 gz8 = {0, 0, 0, 0, 0, 0, 0, 0};     // extra group (clang-23 form)
        __builtin_amdgcn_tensor_load_to_lds(g0, g1, gz4, gz4, gz8, 0);
        __builtin_amdgcn_s_wait_tensorcnt(0);
    }
#else
    // --- fallback: cooperative float4 staging ---
    {
        const float4* Xg = (const float4*)(X + (long)row0 * IN_DIM);
        float4* As4 = (float4*)As;
#pragma unroll
        for (int i = 0; i < 8; ++i) {
            int f = t + i * 128;
            int r = f >> 6;
            int c4 = f & 63;
            As4[r * (APAD / 4) + c4] = Xg[f];
        }
    }
#endif
    __syncthreads();

    const int wave = t >> 5;          // 0..3 -> column tile
    const int lane = t & 31;
    const int m = lane & 15;          // row within tile / n within B,C
    const int half = lane >> 4;       // 0 or 1
    const int col = wave * 16 + m;

    v8f c = {};
#pragma unroll 4
    for (int kb = 0; kb < IN_DIM; kb += 4) {
        const int k0 = kb + half * 2;
        v2f a, b;
        a.x = As[m * APAD + k0];
        a.y = As[m * APAD + k0 + 1];
        b.x = W[k0 * HID + col];
        b.y = W[(k0 + 1) * HID + col];
        c = __builtin_amdgcn_wmma_f32_16x16x4_f32(false, a, false, b,
                                                  (short)0, c, false, false);
    }
#pragma unroll
    for (int v = 0; v < 8; ++v) {
        int mr = v + 8 * half;
        Hw[(long)(row0 + mr) * HID + col] = c[v];
    }
}

// ---------------------------------------------------------------------------
// GEMM1: Hw1[N,16] = ELU(Agg[N,64]) @ W1[64,16], ELU fused into A fragments.
// block = 128 (4 waves), one 16-row tile per wave. Early-exit is wave-uniform.
// ---------------------------------------------------------------------------
__global__ void __launch_bounds__(128) gemm1_wmma(const float* __restrict__ Agg,
                                                  const float* __restrict__ W1,
                                                  float* __restrict__ Hw1,
                                                  int ntiles) {
    const int tile = blockIdx.x * 4 + (threadIdx.x >> 5);  // wave-uniform
    if (tile >= ntiles) return;
    const int lane = threadIdx.x & 31;
    const int m = lane & 15;
    const int half = lane >> 4;
    const int row0 = tile * 16;

    v8f c = {};
#pragma unroll
    for (int kb = 0; kb < HID; kb += 4) {
        const int k0 = kb + half * 2;
        v2f a, b;
        float x0 = Agg[(long)(row0 + m) * HID + k0];
        float x1 = Agg[(long)(row0 + m) * HID + k0 + 1];
        a.x = x0 > 0.0f ? x0 : expm1f(x0);   // fused ELU
        a.y = x1 > 0.0f ? x1 : expm1f(x1);
        b.x = W1[k0 * OUTD + m];
        b.y = W1[(k0 + 1) * OUTD + m];
        c = __builtin_amdgcn_wmma_f32_16x16x4_f32(false, a, false, b,
                                                  (short)0, c, false, false);
    }
#pragma unroll
    for (int v = 0; v < 8; ++v)
        Hw1[(long)(row0 + v + 8 * half) * OUTD + m] = c[v];
}

// ---------------------------------------------------------------------------
// pr[i] = Hw[i,:] . a[0:d] ; pc[i] = Hw[i,:] . a[d:2d]
// ---------------------------------------------------------------------------
__global__ void proj_kernel(const float* __restrict__ Hw,
                            const float* __restrict__ a, int n, int d,
                            float* __restrict__ pr, float* __restrict__ pc) {
    int i = blockIdx.x * blockDim.x + threadIdx.x;
    if (i >= n) return;
    float sr = 0.0f, sc = 0.0f;
    for (int j = 0; j < d; ++j) {
        float h = Hw[(long)i * d + j];
        sr += h * a[j];
        sc += h * a[d + j];
    }
    pr[i] = sr;
    pc[i] = sc;
}

// ---------------------------------------------------------------------------
// edge pass 1: s = leaky_relu(pr[row]+pc[col]); store s; segment-max into
// smax_u[row] via monotone-uint atomicMax.
// ---------------------------------------------------------------------------
__global__ void edge_max_kernel(const int* __restrict__ row,
                                const int* __restrict__ col,
                                const float* __restrict__ pr,
                                const float* __restrict__ pc,
                                float* __restrict__ escore,
                                unsigned* __restrict__ smax_u, int ne) {
    int e = blockIdx.x * blockDim.x + threadIdx.x;
    if (e >= ne) return;
    int r = row[e], cl = col[e];
    float s = pr[r] + pc[cl];
    s = (s >= 0.0f) ? s : NEG_SLOPE * s;
    escore[e] = s;
    atomicMax(&smax_u[r], enc_f(s));
}

// ---------------------------------------------------------------------------
// edge pass 2: ex = exp(s - smax[row]); store ex; segment-sum into ssum[row].
// ---------------------------------------------------------------------------
__global__ void edge_expsum_kernel(const int* __restrict__ row,
                                   float* __restrict__ escore,
                                   const unsigned* __restrict__ smax_u,
                                   float* __restrict__ ssum, int ne) {
    int e = blockIdx.x * blockDim.x + threadIdx.x;
    if (e >= ne) return;
    int r = row[e];
    float ex = expf(escore[e] - dec_f(smax_u[r]));
    escore[e] = ex;
    atomicAdd(&ssum[r], ex);
}

// ---------------------------------------------------------------------------
// edge pass 3 (d=64): out[row,:] += (ex/ssum[row]) * Hsrc[col,:]
// one wave per edge; lane handles features {lane, lane+32} (coalesced 256B
// gather per edge, L2-resident).
// ---------------------------------------------------------------------------
__global__ void edge_aggr64_kernel(const int* __restrict__ row,
                                   const int* __restrict__ col,
                                   const float* __restrict__ escore,
                                   const float* __restrict__ ssum,
                                   const float* __restrict__ Hsrc,
                                   float* __restrict__ out, int ne) {
    int e = (int)((blockIdx.x * (long)blockDim.x + threadIdx.x) >> 5);
    int lane = threadIdx.x & 31;
    if (e >= ne) return;
    int r = row[e], cl = col[e];
    float att = escore[e] / ssum[r];
    const float* h = Hsrc + (long)cl * HID;
    atomicAdd(&out[(long)r * HID + lane], att * h[lane]);
    atomicAdd(&out[(long)r * HID + lane + 32], att * h[lane + 32]);
}

// edge pass 3 (d=16): 16 lanes per edge.
__global__ void edge_aggr16_kernel(const int* __restrict__ row,
                                   const int* __restrict__ col,
                                   const float* __restrict__ escore,
                                   const float* __restrict__ ssum,
                                   const float* __restrict__ Hsrc,
                                   float* __restrict__ out, int ne) {
    int e = (int)((blockIdx.x * (long)blockDim.x + threadIdx.x) >> 4);
    int lane = threadIdx.x & 15;
    if (e >= ne) return;
    int r = row[e], cl = col[e];
    float att = escore[e] / ssum[r];
    atomicAdd(&out[(long)r * OUTD + lane], att * Hsrc[(long)cl * OUTD + lane]);
}

// ---------------------------------------------------------------------------
// in-place log_softmax over 16 columns
// ---------------------------------------------------------------------------
__global__ void logsoftmax16_kernel(float* __restrict__ out, int n) {
    int i = blockIdx.x * blockDim.x + threadIdx.x;
    if (i >= n) return;
    float v[OUTD];
    float mx = -3.402823466e38f;
#pragma unroll
    for (int j = 0; j < OUTD; ++j) {
        v[j] = out[(long)i * OUTD + j];
        mx = fmaxf(mx, v[j]);
    }
    float s = 0.0f;
#pragma unroll
    for (int j = 0; j < OUTD; ++j) s += expf(v[j] - mx);
    float lse = mx + logf(s);
#pragma unroll
    for (int j = 0; j < OUTD; ++j) out[(long)i * OUTD + j] = v[j] - lse;
}

// ---------------------------------------------------------------------------
// launcher
// ---------------------------------------------------------------------------
extern "C" void kernel_launch(void* const* d_in, const int* in_sizes, int n_in,
                              void* d_out, int out_size, void* d_ws, size_t ws_size,
                              hipStream_t stream) {
    const float* X   = (const float*)d_in[0];          // [N,256]
    const int*   ei  = (const int*)d_in[1];            // [2,E] flat
    const float* W0  = (const float*)d_in[2];          // [256,64]
    const float* a0  = (const float*)d_in[3];          // [128]
    const float* W1  = (const float*)d_in[4];          // [64,16]
    const float* a1  = (const float*)d_in[5];          // [32]
    float* out = (float*)d_out;                        // [N,16]

    const int* row = ei;
    const int* col = ei + N_EDGES;

    // workspace layout (floats)
    float* ws      = (float*)d_ws;
    float* Hw0     = ws;                               // N*64
    float* Agg0    = Hw0 + (long)N_NODES * HID;        // N*64
    float* Hw1     = Agg0 + (long)N_NODES * HID;       // N*16
    float* pr      = Hw1 + (long)N_NODES * OUTD;       // N
    float* pc      = pr + N_NODES;                     // N
    unsigned* smax = (unsigned*)(pc + N_NODES);        // N (uint)
    float* ssum    = (float*)smax + N_NODES;           // N
    float* escore  = ssum + N_NODES;                   // E

    const int BT = 256;
    const int gN      = (N_NODES + BT - 1) / BT;
    const int gE      = (N_EDGES + BT - 1) / BT;
    const int g2N     = (2 * N_NODES + BT - 1) / BT;
    const int gAgg0   = (N_NODES * HID + BT - 1) / BT;
    const int gOut    = (N_NODES * OUTD + BT - 1) / BT;
    const int gAgg64  = (N_EDGES + 7) / 8;     // 1 wave / edge, 8 waves / block
    const int gAgg16  = (N_EDGES + 15) / 16;   // 16 lanes / edge
    const int ntiles  = N_NODES / 16;          // 3125 exactly

    // ---- layer 0 ----
    gemm0_wmma<<<ntiles, 128, 0, stream>>>(X, W0, Hw0);
    proj_kernel<<<gN, BT, 0, stream>>>(Hw0, a0, N_NODES, HID, pr, pc);
    zero_kernel<<<g2N, BT, 0, stream>>>((float*)smax, 2 * N_NODES);  // smax+ssum
    edge_max_kernel<<<gE, BT, 0, stream>>>(row, col, pr, pc, escore, smax, N_EDGES);
    edge_expsum_kernel<<<gE, BT, 0, stream>>>(row, escore, smax, ssum, N_EDGES);
    zero_kernel<<<gAgg0, BT, 0, stream>>>(Agg0, N_NODES * HID);
    edge_aggr64_kernel<<<gAgg64, BT, 0, stream>>>(row, col, escore, ssum, Hw0,
                                                  Agg0, N_EDGES);

    // ---- layer 1 (ELU fused into GEMM1 A-load) ----
    gemm1_wmma<<<(ntiles + 3) / 4, 128, 0, stream>>>(Agg0, W1, Hw1, ntiles);
    proj_kernel<<<gN, BT, 0, stream>>>(Hw1, a1, N_NODES, OUTD, pr, pc);
    zero_kernel<<<g2N, BT, 0, stream>>>((float*)smax, 2 * N_NODES);
    zero_kernel<<<gOut, BT, 0, stream>>>(out, N_NODES * OUTD);
    edge_max_kernel<<<gE, BT, 0, stream>>>(row, col, pr, pc, escore, smax, N_EDGES);
    edge_expsum_kernel<<<gE, BT, 0, stream>>>(row, escore, smax, ssum, N_EDGES);
    edge_aggr16_kernel<<<gAgg16, BT, 0, stream>>>(row, col, escore, ssum, Hw1,
                                                  out, N_EDGES);

    // ---- final log_softmax ----
    logsoftmax16_kernel<<<gN, BT, 0, stream>>>(out, N_NODES);
}